// DynamicConv_18305150615689
// MI455X (gfx1250) — compile-verified
//
#include <hip/hip_runtime.h>
#include <hip/hip_bf16.h>

// ---------------------------------------------------------------------------
// DynamicConv for MI455X (gfx1250, wave32)
//   B=32, C=256, H=W=56, K=3
//   k1: per-(b,c) plane mean            -> pooled[32,256]        (ws)
//   k2: WMMA f32 16x16x4 GEMM+bias+relu -> kern[32,2304]         (ws)
//   k3: async-to-LDS tiled dynamic 3x3 depthwise -> out[32,256,56,56]
// Memory-bound: ~206MB effective HBM traffic (x stays resident in 192MB L2
// between k1 and k3), compute is negligible.
// ---------------------------------------------------------------------------

#define BB   32
#define CC   256
#define HH   56
#define WW   56
#define HW   (HH * WW)        // 3136
#define CKK  (CC * 9)         // 2304
#define KDIM CC               // 256 (GEMM reduction dim)

typedef __attribute__((ext_vector_type(2))) float v2f;
typedef __attribute__((ext_vector_type(8))) float v8f;

// ---------------------------------------------------------------------------
// Kernel 1: global average pool over each (b,c) 56x56 plane.
// One 256-thread block (8 waves) per plane; float4 (B128) loads; LDS tree
// reduction. Default RT temporal hint keeps x resident in L2 for kernel 3.
// ---------------------------------------------------------------------------
__global__ void dynconv_pool(const float* __restrict__ x,
                             float* __restrict__ pooled) {
  const int bc  = blockIdx.x;            // 0..8191
  const int tid = threadIdx.x;           // 0..255
  const float4* p4 = (const float4*)(x + (size_t)bc * HW);

  float s = 0.f;
  for (int i = tid; i < HW / 4; i += 256) {   // 784 float4 per plane
    float4 v = p4[i];
    s += v.x + v.y + v.z + v.w;
  }

  __shared__ float red[256];
  red[tid] = s;
  __syncthreads();
  for (int step = 128; step > 0; step >>= 1) {
    if (tid < step) red[tid] += red[tid + step];
    __syncthreads();
  }
  if (tid == 0) pooled[bc] = red[0] * (1.0f / (float)HW);
}

// ---------------------------------------------------------------------------
// Kernel 2: kern = relu(pooled @ Wk^T + bk)  via V_WMMA_F32_16X16X4_F32.
//   pooled : [32, 256]   row-major (A)
//   Wk     : [2304, 256] row-major  => B(k,n) = Wk[n*256 + k]
//   kern   : [32, 2304]
// One wave (32 threads) per 16x16 D tile; grid = (144 N-tiles, 2 M-tiles).
// ISA f32 16x4 A layout: lanes 0-15 carry M=0..15 with {K0,K1}; lanes 16-31
// carry M=0..15 with {K2,K3}. B mirrored with lane = N. D: 8 VGPRs, lane
// halves carry M=j / M=j+8, N = lane&15.
// ---------------------------------------------------------------------------
__global__ void dynconv_gemm(const float* __restrict__ pooled,
                             const float* __restrict__ Wk,
                             const float* __restrict__ bk,
                             float* __restrict__ kern) {
  const int n0   = blockIdx.x * 16;      // 0..2288
  const int m0   = blockIdx.y * 16;      // 0 or 16
  const int lane = threadIdx.x;          // 0..31 (wave32, EXEC all ones)
  const int half = lane >> 4;            // 0: K pair {k,k+1}; 1: {k+2,k+3}
  const int l    = lane & 15;

  v8f acc = {};
  for (int k = 0; k < KDIM; k += 4) {
    const int ka = k + 2 * half;
    // A fragment: rows m0+l, cols {ka, ka+1} (8B-aligned pair loads)
    v2f a = *(const v2f*)(pooled + (m0 + l) * KDIM + ka);
    // B fragment: B(ka..ka+1, n0+l) = Wk[(n0+l)*256 + ka..ka+1]
    v2f b = *(const v2f*)(Wk + (size_t)(n0 + l) * KDIM + ka);
    // D = A*B + C  (8-arg form: neg_a, A, neg_b, B, c_mod, C, reuse_a, reuse_b)
    acc = __builtin_amdgcn_wmma_f32_16x16x4_f32(
        false, a, false, b, (short)0, acc, false, false);
  }

  const float bias = bk[n0 + l];
#pragma unroll
  for (int j = 0; j < 8; ++j) {
    const int row = m0 + j + half * 8;
    float v = acc[j] + bias;
    kern[(size_t)row * CKK + (n0 + l)] = v > 0.f ? v : 0.f;
  }
}

// ---------------------------------------------------------------------------
// Kernel 3: dynamic depthwise 3x3 conv, padding 1.
// One 256-thread block per (b,c) plane. The 12.5KB plane is staged in LDS via
// CDNA5 async global->LDS B128 movers (ASYNCcnt-tracked; data never touches
// VGPRs), completed with s_wait_asynccnt before the workgroup barrier. Each x
// element crosses global->LDS exactly once (mostly an L2 hit thanks to
// kernel 1) and is reused 9x out of LDS.
// ---------------------------------------------------------------------------
__global__ void dynconv_dw(const float* __restrict__ x,
                           const float* __restrict__ kern,
                           float* __restrict__ out) {
  const int bc  = blockIdx.x;            // 0..8191
  const int tid = threadIdx.x;           // 0..255

  __shared__ __align__(16) float plane[HW];
  __shared__ float w[16];

  {
    const float* src = x + (size_t)bc * HW;
    // Low 32 bits of a generic pointer to LDS == LDS byte offset (ISA 10.2).
    const unsigned lds_base = (unsigned)(unsigned long long)(void*)plane;
    for (int i = tid; i < HW / 4; i += 256) {   // 784 B128 transfers
      const unsigned lds_addr = lds_base + (unsigned)(i * 16);
      const float* gaddr = src + i * 4;
      asm volatile("global_load_async_to_lds_b128 %0, %1, off"
                   :: "v"(lds_addr), "v"(gaddr)
                   : "memory");
    }
  }
  if (tid < 9) w[tid] = kern[(size_t)bc * 9 + tid];
  asm volatile("s_wait_asynccnt 0x0" ::: "memory");
  __syncthreads();

  float* dst = out + (size_t)bc * HW;
  for (int p = tid; p < HW; p += 256) {
    const int y  = p / WW;
    const int xx = p - y * WW;
    float s = 0.f;
#pragma unroll
    for (int dy = -1; dy <= 1; ++dy) {
      const int yy = y + dy;
      if (yy < 0 || yy >= HH) continue;
#pragma unroll
      for (int dx = -1; dx <= 1; ++dx) {
        const int xc = xx + dx;
        if (xc < 0 || xc >= WW) continue;
        s += w[(dy + 1) * 3 + (dx + 1)] * plane[yy * WW + xc];
      }
    }
    dst[p] = s;
  }
}

// ---------------------------------------------------------------------------
extern "C" void kernel_launch(void* const* d_in, const int* in_sizes, int n_in,
                              void* d_out, int out_size, void* d_ws, size_t ws_size,
                              hipStream_t stream) {
  const float* x  = (const float*)d_in[0];   // [32,256,56,56]
  const float* Wk = (const float*)d_in[1];   // [2304,256]
  const float* bk = (const float*)d_in[2];   // [2304]
  float* out = (float*)d_out;                // [32,256,56,56]

  float* pooled = (float*)d_ws;              // 8192 floats
  float* kern   = pooled + BB * CC;          // 73728 floats

  // k1: mean pool (one block per (b,c) plane)
  dynconv_pool<<<dim3(BB * CC), dim3(256), 0, stream>>>(x, pooled);

  // k2: WMMA generator GEMM + bias + relu (one wave per 16x16 tile)
  dynconv_gemm<<<dim3(CKK / 16, BB / 16), dim3(32), 0, stream>>>(pooled, Wk, bk, kern);

  // k3: dynamic depthwise conv, async-to-LDS staging (one block per plane)
  dynconv_dw<<<dim3(BB * CC), dim3(256), 0, stream>>>(x, kern, out);
}